// ConProximity_52613349376866
// MI455X (gfx1250) — compile-verified
//
#include <hip/hip_runtime.h>
#include <hip/hip_bf16.h>

#define AS1 __attribute__((address_space(1)))
#define AS3 __attribute__((address_space(3)))

typedef int v4i __attribute__((vector_size(16)));   // 16-byte vector pointee for b128 async copy

#if defined(__has_builtin)
#if __has_builtin(__builtin_amdgcn_global_load_async_to_lds_b128) && \
    __has_builtin(__builtin_amdgcn_s_wait_asynccnt)
#define USE_ASYNC_LDS 1
#endif
#if __has_builtin(__builtin_amdgcn_wmma_f32_16x16x4_f32)
#define USE_WMMA_REDUCE 1
#endif
#endif

namespace {
constexpr int kB = 2048;
constexpr int kC = 100;
constexpr int kD = 1024;
constexpr int kCT = 10;                 // centers per LDS tile (10 * 4KB = 40KB)
constexpr int kRows = 8;                // one row per wave, 8 waves / block
constexpr int kTiles = kC / kCT;        // 10
constexpr int kBlocks = kB / kRows;     // 256 partials -> exactly 4x (4x16) WMMA chunks
constexpr int kThreads = kRows * 32;    // 256 threads (wave32!)
}

typedef __attribute__((ext_vector_type(2))) float v2f;
typedef __attribute__((ext_vector_type(8))) float v8f;

// ---------------------------------------------------------------------------
// Kernel 1: per-block partial masked sums of tropical distances.
// x rows resident in VGPRs; centers staged through LDS tiles via async copy.
// ---------------------------------------------------------------------------
__global__ __launch_bounds__(kThreads)
void trop_partial_kernel(const float* __restrict__ x,
                         const int* __restrict__ labels,
                         const float* __restrict__ centers,
                         float* __restrict__ partial) {
  __shared__ float cs[kCT * kD];   // 40 KB center tile
  __shared__ float wsum[kRows];

  const int lane = threadIdx.x & 31;
  const int wave = threadIdx.x >> 5;
  const int row  = blockIdx.x * kRows + wave;

  // Load this wave's x row into registers: lane holds 8 float4 = 32 elements.
  float4 xr[8];
  const float4* xp4 = (const float4*)(x + (size_t)row * kD);
#pragma unroll
  for (int k = 0; k < 8; ++k) xr[k] = xp4[k * 32 + lane];

  const int lab = labels[row];
  float acc = 0.0f;

  constexpr int kVec4PerTile = (kCT * kD) / 4;          // 2560 float4
  constexpr int kCopyIters   = kVec4PerTile / kThreads; // 10 per thread

  for (int t = 0; t < kTiles; ++t) {
    __syncthreads();  // previous tile fully consumed before overwrite
    const float* src = centers + (size_t)t * kCT * kD;
#if USE_ASYNC_LDS
#pragma unroll
    for (int i = 0; i < kCopyIters; ++i) {
      const int idx = threadIdx.x + kThreads * i;       // float4 index
      __builtin_amdgcn_global_load_async_to_lds_b128(
          (AS1 v4i*)(src + (size_t)idx * 4),
          (AS3 v4i*)(&cs[idx * 4]),
          0, 0);
    }
    __builtin_amdgcn_s_wait_asynccnt(0);
#else
#pragma unroll
    for (int i = 0; i < kCopyIters; ++i) {
      const int idx = threadIdx.x + kThreads * i;
      ((float4*)cs)[idx] = ((const float4*)src)[idx];
    }
#endif
    if (t + 1 < kTiles) {
      // Warm L2/WGP$ for the next tile (gfx1250 global_prefetch_b8).
      __builtin_prefetch(centers + (size_t)(t + 1) * kCT * kD + threadIdx.x * 40, 0, 0);
    }
    __syncthreads();

    for (int c = 0; c < kCT; ++c) {
      const float4* cp4 = (const float4*)(cs + c * kD);
      float mx = -__builtin_inff();
      float mn =  __builtin_inff();
#pragma unroll
      for (int k = 0; k < 8; ++k) {
        const float4 cv = cp4[k * 32 + lane];
        const float d0 = xr[k].x - cv.x;
        const float d1 = xr[k].y - cv.y;
        const float d2 = xr[k].z - cv.z;
        const float d3 = xr[k].w - cv.w;
        mx = fmaxf(fmaxf(fmaxf(mx, d0), fmaxf(d1, d2)), d3);
        mn = fminf(fminf(fminf(mn, d0), fminf(d1, d2)), d3);
      }
      // wave32 butterfly reduce (all lanes end with full max/min)
#pragma unroll
      for (int off = 16; off > 0; off >>= 1) {
        mx = fmaxf(mx, __shfl_xor(mx, off, 32));
        mn = fminf(mn, __shfl_xor(mn, off, 32));
      }
      if (t * kCT + c != lab) acc += (mx - mn);
    }
  }

  if (lane == 0) wsum[wave] = acc;
  __syncthreads();
  if (threadIdx.x == 0) {
    float s = 0.0f;
#pragma unroll
    for (int w = 0; w < kRows; ++w) s += wsum[w];
    partial[blockIdx.x] = s;
  }
}

// ---------------------------------------------------------------------------
// Kernel 2: deterministic reduction of 256 partials -> scalar mean.
// Uses v_wmma_f32_16x16x4_f32 with an all-ones A matrix: 4 accumulating WMMAs
// produce per-column sums of the 4x(4x16) partial chunks; a 4-step shuffle
// finishes. A==ones makes the result independent of the exact K-slot layout.
// ---------------------------------------------------------------------------
__global__ __launch_bounds__(32)
void trop_finalize_kernel(const float* __restrict__ partial,
                          float* __restrict__ out) {
  const int lane = threadIdx.x;  // single wave32, EXEC all ones (no divergence)
  float v;
#if USE_WMMA_REDUCE
  v2f a; a[0] = 1.0f; a[1] = 1.0f;   // 16x4 all-ones A
  v8f c = {};
  const int n  = lane & 15;
  const int hi = lane >> 4;
#pragma unroll
  for (int j = 0; j < 4; ++j) {
    v2f b;
    b[0] = partial[j * 64 + (hi * 2 + 0) * 16 + n];
    b[1] = partial[j * 64 + (hi * 2 + 1) * 16 + n];
    c = __builtin_amdgcn_wmma_f32_16x16x4_f32(false, a, false, b, (short)0, c,
                                              false, false);
  }
  v = c[0];  // column sum for this lane's n (both halves hold identical rows)
#pragma unroll
  for (int off = 8; off > 0; off >>= 1) v += __shfl_xor(v, off, 32);
#else
  v = 0.0f;
#pragma unroll
  for (int j = 0; j < 8; ++j) v += partial[lane * 8 + j];
#pragma unroll
  for (int off = 16; off > 0; off >>= 1) v += __shfl_xor(v, off, 32);
#endif
  if (lane == 0) out[0] = v * (1.0f / ((float)kB * (float)(kC - 1)));
}

extern "C" void kernel_launch(void* const* d_in, const int* in_sizes, int n_in,
                              void* d_out, int out_size, void* d_ws, size_t ws_size,
                              hipStream_t stream) {
  (void)in_sizes; (void)n_in; (void)out_size; (void)ws_size;
  const float* x       = (const float*)d_in[0];
  const int*   labels  = (const int*)d_in[1];
  const float* centers = (const float*)d_in[2];
  float* partial = (float*)d_ws;   // 256 floats of scratch

  trop_partial_kernel<<<kBlocks, kThreads, 0, stream>>>(x, labels, centers, partial);
  trop_finalize_kernel<<<1, 32, 0, stream>>>(partial, (float*)d_out);
}